// Hebbian_56246891708778
// MI455X (gfx1250) — compile-verified
//
#include <hip/hip_runtime.h>

typedef __attribute__((ext_vector_type(2))) float v2f;
typedef __attribute__((ext_vector_type(8))) float v8f;

#define SDIM 128
#define LDST 129   // LDS row stride (floats): 129 % 64 == 1 -> conflict-free rows AND columns

__device__ __forceinline__ v8f wmma4(v2f a, v2f b, v8f c) {
  // D = A(16x4 f32) * B(4x16 f32) + C(16x16 f32)
  return __builtin_amdgcn_wmma_f32_16x16x4_f32(
      /*neg_a=*/false, a, /*neg_b=*/false, b,
      /*c_mod=*/(short)0, c, /*reuse_a=*/false, /*reuse_b=*/false);
}

__global__ __launch_bounds__(256)
void hebbian_kernel(const float* __restrict__ obs,
                    const float* __restrict__ brim,
                    const float* __restrict__ modulation,
                    const float* __restrict__ w_assoc,
                    const float* __restrict__ key_W,
                    const float* __restrict__ key_b,
                    const float* __restrict__ val_W,
                    const float* __restrict__ val_b,
                    const float* __restrict__ A_w,
                    const float* __restrict__ B_w,
                    const int*   __restrict__ done,
                    float* __restrict__ out)
{
  extern __shared__ float smem[];
  float* sK   = smem;                    // 128 x 129 floats
  float* sV   = smem + SDIM * LDST;      // 128 x 129 floats
  float* sMod = smem + 2 * SDIM * LDST;  // 128 floats

  const int b    = blockIdx.x;
  const int tid  = threadIdx.x;
  const int lane = tid & 31;
  const int wid  = tid >> 5;     // 0..7 : row-block of this wave
  const int l16  = lane & 15;    // N / M index within tile
  const int lh   = lane >> 4;    // half select: K offset 0 or 2

  if (tid < SDIM) sMod[tid] = modulation[(size_t)b * SDIM + tid];
  __syncthreads();

  const size_t bOff  = (size_t)b * SDIM * SDIM;
  const float* obsB  = obs  + bOff;
  const float* brimB = brim + bOff;

  // ------------- Phase 1a: k = obs @ key_W^T + key_b  -> sK -------------
  {
    v8f acc[8];
#pragma unroll
    for (int nt = 0; nt < 8; ++nt) acc[nt] = {};
    const int arow = wid * 16 + l16;
    for (int kk = 0; kk < SDIM; kk += 4) {
      const int c0 = kk + lh * 2;
      v2f a;
      a.x = obsB[arow * SDIM + c0];
      a.y = obsB[arow * SDIM + c0 + 1];
#pragma unroll
      for (int nt = 0; nt < 8; ++nt) {
        const int o = nt * 16 + l16;          // B[c,o] = key_W[o,c]
        v2f bf;
        bf.x = key_W[o * SDIM + c0];
        bf.y = key_W[o * SDIM + c0 + 1];
        acc[nt] = wmma4(a, bf, acc[nt]);
      }
    }
#pragma unroll
    for (int nt = 0; nt < 8; ++nt) {
      const float bias = key_b[nt * 16 + l16];
#pragma unroll
      for (int r = 0; r < 8; ++r) {
        const int row = wid * 16 + r + lh * 8;
        sK[row * LDST + nt * 16 + l16] = acc[nt][r] + bias;
      }
    }
  }

  // -------- Phase 1b: v = mod * (brim @ val_W^T + val_b) -> sV --------
  {
    v8f acc[8];
#pragma unroll
    for (int nt = 0; nt < 8; ++nt) acc[nt] = {};
    const int arow = wid * 16 + l16;
    for (int kk = 0; kk < SDIM; kk += 4) {
      const int c0 = kk + lh * 2;
      v2f a;
      a.x = brimB[arow * SDIM + c0];
      a.y = brimB[arow * SDIM + c0 + 1];
#pragma unroll
      for (int nt = 0; nt < 8; ++nt) {
        const int o = nt * 16 + l16;
        v2f bf;
        bf.x = val_W[o * SDIM + c0];
        bf.y = val_W[o * SDIM + c0 + 1];
        acc[nt] = wmma4(a, bf, acc[nt]);
      }
    }
#pragma unroll
    for (int nt = 0; nt < 8; ++nt) {
      const float bias = val_b[nt * 16 + l16];
#pragma unroll
      for (int r = 0; r < 8; ++r) {
        const int row = wid * 16 + r + lh * 8;
        sV[row * LDST + nt * 16 + l16] = (acc[nt][r] + bias) * sMod[row];
      }
    }
  }

  __syncthreads();

  // --- Phase 2: corr = k^T v ; reg = k k ; fused Hebbian update ---
  const bool dn = done[b] != 0;
  const int  i0 = wid * 16 + l16;          // output row for A fragments
#pragma unroll 2
  for (int nt = 0; nt < 8; ++nt) {
    v8f accC = {};   // corr tile
    v8f accR = {};   // reg tile
    const int j = nt * 16 + l16;           // output col for B fragments
    for (int kk = 0; kk < SDIM; kk += 4) {
      const int c0 = kk + lh * 2;
      v2f aC;  // corr A[i,s] = k[s,i]  (column read of sK)
      aC.x = sK[c0 * LDST + i0];
      aC.y = sK[(c0 + 1) * LDST + i0];
      v2f aR;  // reg  A[i,m] = k[i,m]  (row read of sK)
      aR.x = sK[i0 * LDST + c0];
      aR.y = sK[i0 * LDST + c0 + 1];
      v2f bC;  // corr B[s,j] = v[s,j]
      bC.x = sV[c0 * LDST + j];
      bC.y = sV[(c0 + 1) * LDST + j];
      v2f bR;  // reg  B[m,j] = k[m,j]
      bR.x = sK[c0 * LDST + j];
      bR.y = sK[(c0 + 1) * LDST + j];
      accC = wmma4(aC, bC, accC);
      accR = wmma4(aR, bR, accR);
    }
#pragma unroll
    for (int r = 0; r < 8; ++r) {
      const int gi = wid * 16 + r + lh * 8;
      const size_t idx = bOff + (size_t)gi * SDIM + j;
      const float w  = w_assoc[idx];
      const float aw = A_w[gi * SDIM + j];
      const float bw = B_w[gi * SDIM + j];
      const float res = w + aw * (1.0f - w) * accC[r] - bw * w * accR[r];
      out[idx] = dn ? res : w;
    }
  }
}

extern "C" void kernel_launch(void* const* d_in, const int* in_sizes, int n_in,
                              void* d_out, int out_size, void* d_ws, size_t ws_size,
                              hipStream_t stream) {
  const float* obs        = (const float*)d_in[0];
  const float* brim       = (const float*)d_in[1];
  const float* modulation = (const float*)d_in[2];
  const float* w_assoc    = (const float*)d_in[3];
  const float* key_W      = (const float*)d_in[4];
  const float* key_b      = (const float*)d_in[5];
  const float* val_W      = (const float*)d_in[6];
  const float* val_b      = (const float*)d_in[7];
  const float* A_w        = (const float*)d_in[8];
  const float* B_w        = (const float*)d_in[9];
  const int*   done       = (const int*)d_in[10];
  float* out = (float*)d_out;

  const int Bn = in_sizes[0] / (SDIM * SDIM);           // 2048
  const size_t shmem = (size_t)(2 * SDIM * LDST + SDIM) * sizeof(float);  // ~130 KB

  hebbian_kernel<<<Bn, 256, shmem, stream>>>(
      obs, brim, modulation, w_assoc, key_W, key_b, val_W, val_b,
      A_w, B_w, done, out);
}